// HNN_39968965657036
// MI455X (gfx1250) — compile-verified
//
#include <hip/hip_runtime.h>

// HNN leapfrog on MI455X (gfx1250): persistent per-row-tile kernel,
// f32 WMMA (V_WMMA_F32_16X16X4_F32) for exact-f32 GEMMs.
// Round 4: revert to the Round-2 K-loop (compiler-scheduled, no manual
// pipelining -> no spills, compact 64-WMMA static footprint), keep the
// paired epilogue addressing (ds_*_2addr-friendly).

#define DD        256
#define TWO_D     512
#define STEPS     16
#define DT        0.1f
#define ROWS      32                 // batch rows per workgroup (2 M-tiles)
#define NWAVES    16
#define NTHREADS  (NWAVES * 32)
#define LDS_STRIDE (TWO_D + 4)       // pad to avoid 64-bank conflicts

typedef __attribute__((ext_vector_type(2))) float v2f;
typedef __attribute__((ext_vector_type(8))) float v8f;

#define WMMA_F32(A, B, C) \
  __builtin_amdgcn_wmma_f32_16x16x4_f32(false, (A), false, (B), (short)0, (C), false, false)

// ---------------------------------------------------------------------------
// Pack a 512x512 weight matrix into WMMA B-operand pair order.
// Logical B[k][n] = transposed ? W[n][k] : W[k][n].
// Dest: P[((k>>2)*2 + ((k>>1)&1)) * 1024 + n*2 + (k&1)]
// so that lane (khalf = lane>=16, n) loads the b64 pair
// {B[k0+2*khalf][n], B[k0+2*khalf+1][n]} with one global_load_b64.
// ---------------------------------------------------------------------------
__global__ __launch_bounds__(256) void pack_weights(const float* __restrict__ W,
                                                    float* __restrict__ P,
                                                    const int transposed) {
  const int idx = blockIdx.x * 256 + threadIdx.x;   // 0 .. 512*512-1
  const int n = idx & 511;
  const int k = idx >> 9;
  const float v = transposed ? W[n * TWO_D + k] : W[k * TWO_D + n];
  const int grp = ((k >> 2) << 1) + ((k >> 1) & 1);
  P[grp * (TWO_D * 2) + (n << 1) + (k & 1)] = v;
}

// ---------------------------------------------------------------------------
// One (32 x 512) = (32 x 512) @ (512 x 512) GEMM using f32 WMMA.
// Each wave: 2 M-tiles x 2 N-tiles (columns [wave*32, wave*32+32)).
// MODE 1: +b1, relu          (sCst[0..511])
// MODE 2: +b2, (z>0)?Wo:0    (sCst[512..1023] bias, sCst[1024..1535] Wo)
// MODE 3: mask by (Maskbuf>0); Obuf may alias Maskbuf (read-then-write per lane)
// MODE 4: plain store
// ---------------------------------------------------------------------------
template <int MODE>
__device__ __forceinline__ void gemm_512(const float* Abuf,
                                         const float* __restrict__ Bpk,
                                         float* Obuf,
                                         const float* Maskbuf,
                                         const float* __restrict__ sCst,
                                         int wave, int lane) {
  const int m     = lane & 15;
  const int khalf = lane >> 4;                 // 0 or 1 (K shift of 2)

  v8f acc[2][2];
#pragma unroll
  for (int mt = 0; mt < 2; ++mt)
#pragma unroll
    for (int t = 0; t < 2; ++t)
#pragma unroll
      for (int j = 0; j < 8; ++j) acc[mt][t][j] = 0.0f;

  const float* A0 = Abuf + m * LDS_STRIDE;          // M-tile 0, row m
  const float* A1 = A0 + 16 * LDS_STRIDE;           // M-tile 1
  // Per-lane packed-B base: pair (k0,khalf,n) at Bpk[((k0>>1)+khalf)*1024 + n*2]
  const float* Bbase = Bpk + khalf * (TWO_D * 2) + (wave * 32 + m) * 2;

#pragma unroll 2
  for (int k0 = 0; k0 < TWO_D; k0 += 4) {
    const int kk = k0 + (khalf << 1);
    // A 16x4 tiles: VGPR0 = K=2*khalf, VGPR1 = K=2*khalf+1 (ISA 32-bit layout)
    v2f a0 = *(const v2f*)(A0 + kk);
    v2f a1 = *(const v2f*)(A1 + kk);
    const float* Bp = Bbase + (k0 >> 1) * (TWO_D * 2);
    v2f b0 = *(const v2f*)(Bp);                 // N-tile 0 (cols wave*32 + 0..15)
    v2f b1 = *(const v2f*)(Bp + 32);            // N-tile 1 (cols wave*32 + 16..31)
    acc[0][0] = WMMA_F32(a0, b0, acc[0][0]);
    acc[0][1] = WMMA_F32(a0, b1, acc[0][1]);
    acc[1][0] = WMMA_F32(a1, b0, acc[1][0]);
    acc[1][1] = WMMA_F32(a1, b1, acc[1][1]);
  }

  // C/D layout: VGPR j -> row (j + 8*khalf) in tile, col = lane&15 in tile.
  // Both N-tiles of a row go through one base pointer (o[0], o[16]) so the
  // compiler can merge into ds_store_2addr_b32 / ds_load_2addr_b32.
  const int rbase = khalf << 3;
  const int colb  = wave * 32 + m;
  const float c0a = (MODE == 1) ? sCst[colb]              : (MODE == 2) ? sCst[TWO_D + colb]      : 0.0f;
  const float c1a = (MODE == 1) ? sCst[colb + 16]         : (MODE == 2) ? sCst[TWO_D + colb + 16] : 0.0f;
  const float w0  = (MODE == 2) ? sCst[2 * TWO_D + colb]      : 0.0f;
  const float w1  = (MODE == 2) ? sCst[2 * TWO_D + colb + 16] : 0.0f;

#pragma unroll
  for (int mt = 0; mt < 2; ++mt) {
#pragma unroll
    for (int j = 0; j < 8; ++j) {
      const int row = mt * 16 + rbase + j;
      float* o = Obuf + row * LDS_STRIDE + colb;
      float v0 = acc[mt][0][j];
      float v1 = acc[mt][1][j];
      if (MODE == 1) {
        o[0]  = fmaxf(v0 + c0a, 0.0f);
        o[16] = fmaxf(v1 + c1a, 0.0f);
      } else if (MODE == 2) {
        o[0]  = (v0 + c0a > 0.0f) ? w0 : 0.0f;
        o[16] = (v1 + c1a > 0.0f) ? w1 : 0.0f;
      } else if (MODE == 3) {
        const float* mk = Maskbuf + row * LDS_STRIDE + colb;
        float h0 = mk[0];
        float h1 = mk[16];
        o[0]  = (h0 > 0.0f) ? v0 : 0.0f;
        o[16] = (h1 > 0.0f) ? v1 : 0.0f;
      } else {
        o[0]  = v0;
        o[16] = v1;
      }
    }
  }
}

// ds = grad_H(s) left in sT2.  sS = state, sT1/sT2 scratch activations.
__device__ __forceinline__ void grad_eval(float* sS, float* sT1, float* sT2,
                                          const float* sCst,
                                          const float* __restrict__ P1f,
                                          const float* __restrict__ P2f,
                                          const float* __restrict__ P2b,
                                          const float* __restrict__ P1b,
                                          int wave, int lane) {
  gemm_512<1>(sS,  P1f, sT1, nullptr, sCst, wave, lane);  // h1 = relu(s@W1^T+b1)
  __syncthreads();
  gemm_512<2>(sT1, P2f, sT2, nullptr, sCst, wave, lane);  // g2 = (z2>0)?Wo:0
  __syncthreads();
  gemm_512<3>(sT2, P2b, sT1, sT1,     sCst, wave, lane);  // g1 = (g2@W2)*(h1>0)
  __syncthreads();
  gemm_512<4>(sT1, P1b, sT2, nullptr, sCst, wave, lane);  // ds = g1@W1
  __syncthreads();
}

// ---------------------------------------------------------------------------
// Persistent HNN kernel: one workgroup integrates ROWS rows for all 16 steps.
// ---------------------------------------------------------------------------
__global__ __launch_bounds__(NTHREADS) void hnn_leapfrog_kernel(
    const float* __restrict__ x,   // (8192, 256, 4)
    const float* __restrict__ b1, const float* __restrict__ b2,
    const float* __restrict__ Wo,
    const float* __restrict__ P1f, const float* __restrict__ P2f,
    const float* __restrict__ P2b, const float* __restrict__ P1b,
    float* __restrict__ out)       // (8192, 512)
{
  __shared__ float sS [ROWS * LDS_STRIDE];   // state  [q | p]
  __shared__ float sT1[ROWS * LDS_STRIDE];   // h1, later g1
  __shared__ float sT2[ROWS * LDS_STRIDE];   // g2, later ds
  __shared__ float sCst[3 * TWO_D];          // b1 | b2 | Wo

  const int tid  = threadIdx.x;
  const int wave = tid >> 5;
  const int lane = tid & 31;
  const int r0   = blockIdx.x * ROWS;

  for (int i = tid; i < TWO_D; i += NTHREADS) {
    sCst[i]             = b1[i];
    sCst[TWO_D + i]     = b2[i];
    sCst[2 * TWO_D + i] = Wo[i];
  }

  // q = x[:,:,3]; p = x[:,:,3] - x[:,:,2]
  for (int i = tid; i < ROWS * DD; i += NTHREADS) {
    const int r = i >> 8;
    const int d = i & 255;
    const float* xp = x + ((r0 + r) * DD + d) * 4;
    const float x3 = xp[3];
    const float x2 = xp[2];
    sS[r * LDS_STRIDE + d]      = x3;        // q
    sS[r * LDS_STRIDE + DD + d] = x3 - x2;   // p
  }
  __syncthreads();

  for (int step = 0; step < STEPS; ++step) {
    grad_eval(sS, sT1, sT2, sCst, P1f, P2f, P2b, P1b, wave, lane);
    // p += 0.5*DT*(-dH/dq);  q += DT*(dH/dp)   (both grads from old state)
    for (int i = tid; i < ROWS * DD; i += NTHREADS) {
      const int r = i >> 8;
      const int d = i & 255;
      const float dHq = sT2[r * LDS_STRIDE + d];
      const float dHp = sT2[r * LDS_STRIDE + DD + d];
      sS[r * LDS_STRIDE + DD + d] -= 0.5f * DT * dHq;
      sS[r * LDS_STRIDE + d]      += DT * dHp;
    }
    __syncthreads();

    grad_eval(sS, sT1, sT2, sCst, P1f, P2f, P2b, P1b, wave, lane);
    // p += 0.5*DT*(-dH/dq)
    for (int i = tid; i < ROWS * DD; i += NTHREADS) {
      const int r = i >> 8;
      const int d = i & 255;
      sS[r * LDS_STRIDE + DD + d] -= 0.5f * DT * sT2[r * LDS_STRIDE + d];
    }
    __syncthreads();
  }

  for (int i = tid; i < ROWS * TWO_D; i += NTHREADS) {
    const int r = i >> 9;
    const int c = i & 511;
    out[(r0 + r) * TWO_D + c] = sS[r * LDS_STRIDE + c];
  }
}

// ---------------------------------------------------------------------------
extern "C" void kernel_launch(void* const* d_in, const int* in_sizes, int n_in,
                              void* d_out, int out_size, void* d_ws, size_t ws_size,
                              hipStream_t stream) {
  const float* x  = (const float*)d_in[0];
  const float* W1 = (const float*)d_in[1];
  const float* b1 = (const float*)d_in[2];
  const float* W2 = (const float*)d_in[3];
  const float* b2 = (const float*)d_in[4];
  const float* Wo = (const float*)d_in[5];
  float* out = (float*)d_out;

  const int MS = TWO_D * TWO_D;                    // 256K floats = 1 MB
  float* P1f = (float*)d_ws;                       // B[k][n] = W1[n][k]
  float* P2f = P1f + MS;                           // B[k][n] = W2[n][k]
  float* P2b = P2f + MS;                           // B[k][n] = W2[k][n]
  float* P1b = P2b + MS;                           // B[k][n] = W1[k][n]  (4 MB ws)

  pack_weights<<<(MS + 255) / 256, 256, 0, stream>>>(W1, P1f, 1);
  pack_weights<<<(MS + 255) / 256, 256, 0, stream>>>(W2, P2f, 1);
  pack_weights<<<(MS + 255) / 256, 256, 0, stream>>>(W2, P2b, 0);
  pack_weights<<<(MS + 255) / 256, 256, 0, stream>>>(W1, P1b, 0);

  const int nblocks = 8192 / ROWS;                 // 256 workgroups
  hnn_leapfrog_kernel<<<nblocks, NTHREADS, 0, stream>>>(
      x, b1, b2, Wo, P1f, P2f, P2b, P1b, out);
}